// ForexLSTM_80393197846576
// MI455X (gfx1250) — compile-verified
//
#include <hip/hip_runtime.h>

typedef __attribute__((ext_vector_type(2))) float v2f;
typedef __attribute__((ext_vector_type(8))) float v8f;

constexpr int kH   = 65;      // hidden size
constexpr int kG   = 260;     // 4*H gate columns
constexpr int kGP  = 272;     // gates padded to 17 tiles of 16
constexpr int kKP  = 68;      // K padded to 17 chunks of 4
constexpr int kB   = 512;     // batch
constexpr int kT   = 1024;    // timesteps
constexpr int kNT  = 17;      // N tiles / K chunks
constexpr int kWAVES = 17;
constexpr int kBLK = kWAVES * 32;  // 544 threads

__device__ __forceinline__ float sigm(float x) {
    return 1.0f / (1.0f + __expf(-x));
}
__device__ __forceinline__ float tanh_fast(float x) {
    float e = __expf(2.0f * x);
    return 1.0f - 2.0f / (e + 1.0f);
}

// ---------------------------------------------------------------------------
// Kernel 1: pre[t][b][g] = sum_k x[b][t][k] * w_ih[g][k] + (b_ih[g]+b_hh[g])
// One block per (t, 16-batch tile); 17 waves, each owning one 16-col N tile.
// fp32 WMMA 16x16x4, A staged in LDS, B (weights) register-resident.
// ---------------------------------------------------------------------------
__global__ __launch_bounds__(kBLK)
void lstm_pre_gemm(const float* __restrict__ x,     // (B, T, H)
                   const float* __restrict__ w_ih,  // (4H, H)
                   const float* __restrict__ b_ih,  // (4H)
                   const float* __restrict__ b_hh,  // (4H)
                   float* __restrict__ pre)         // (T, B, 4H)
{
    __shared__ float shA[16 * kKP];

    const int tid = threadIdx.x;
    const int t   = blockIdx.x >> 5;         // T * (B/16) blocks
    const int b0  = (blockIdx.x & 31) << 4;

    // Stage A tile (16 batches x 68 features, zero-padded) into LDS, coalesced.
    for (int idx = tid; idx < 16 * kKP; idx += kBLK) {
        int m = idx / kKP, k = idx - m * kKP;
        float v = 0.0f;
        if (k < kH)
            v = x[(size_t)(b0 + m) * (size_t)(kT * kH) + (size_t)t * kH + k];
        shA[idx] = v;
    }
    __syncthreads();

    const int wave  = tid >> 5;
    const int lane  = tid & 31;
    const int g0    = wave << 4;
    const int n     = lane & 15;
    const int g     = g0 + n;
    const int khalf = (lane >> 4) << 1;   // 0 or 2 (K sub-offset)
    const int mhi   = (lane >> 4) << 3;   // 0 or 8 (C row base)

    // Register-resident weight tile: B[k][n] = w_ih[g][k]
    v2f wreg[kNT];
#pragma unroll
    for (int kc = 0; kc < kNT; ++kc) {
        int kb = (kc << 2) + khalf;
        float bx = 0.0f, by = 0.0f;
        if (g < kG) {
            if (kb     < kH) bx = w_ih[g * kH + kb];
            if (kb + 1 < kH) by = w_ih[g * kH + kb + 1];
        }
        wreg[kc].x = bx; wreg[kc].y = by;
    }

    v8f acc = {};
#pragma unroll
    for (int kc = 0; kc < kNT; ++kc) {
        int kb = (kc << 2) + khalf;
        v2f a = *(const v2f*)&shA[n * kKP + kb];
        acc = __builtin_amdgcn_wmma_f32_16x16x4_f32(
                  false, a, false, wreg[kc], (short)0, acc, false, false);
    }

    if (g < kG) {
        float bias = b_ih[g] + b_hh[g];
#pragma unroll
        for (int r = 0; r < 8; ++r) {
            int m = r + mhi;
            pre[((size_t)t * kB + b0 + m) * (size_t)kG + g] = acc[r] + bias;
        }
    }
}

// ---------------------------------------------------------------------------
// Kernel 2: the recurrent scan. One block per 16-batch tile (32 blocks).
// 17 waves: wave w computes gate columns [16w, 16w+16) via fp32 WMMA with
// register-resident W_hh slice; h lives in LDS, c lives in registers of the
// elementwise lanes. Two barriers per timestep.
// ---------------------------------------------------------------------------
__global__ __launch_bounds__(kBLK)
void lstm_scan(const float* __restrict__ pre,    // (T, B, 4H)
               const float* __restrict__ w_hh,   // (4H, H)
               float* __restrict__ h_all,        // (B, T, H) or nullptr
               float* __restrict__ h_last)       // (B, H)   or nullptr
{
    __shared__ float shH[16 * kKP];   // h tile, zero-padded cols 65..67
    __shared__ float shG[16 * kGP];   // pre-activation gates

    const int tid   = threadIdx.x;
    const int b0    = blockIdx.x << 4;
    const int wave  = tid >> 5;
    const int lane  = tid & 31;
    const int g0    = wave << 4;
    const int n     = lane & 15;
    const int g     = g0 + n;
    const int khalf = (lane >> 4) << 1;
    const int mhi   = (lane >> 4) << 3;

    for (int idx = tid; idx < 16 * kKP; idx += kBLK) shH[idx] = 0.0f;

    // Register-resident recurrent weights: B[k][n] = w_hh[g][k]
    v2f wreg[kNT];
#pragma unroll
    for (int kc = 0; kc < kNT; ++kc) {
        int kb = (kc << 2) + khalf;
        float bx = 0.0f, by = 0.0f;
        if (g < kG) {
            if (kb     < kH) bx = w_hh[g * kH + kb];
            if (kb + 1 < kH) by = w_hh[g * kH + kb + 1];
        }
        wreg[kc].x = bx; wreg[kc].y = by;
    }

    // Elementwise lane state: each thread owns up to 2 (batch, hidden) cells,
    // cell state c kept in registers across all timesteps.
    const int e0 = tid;           // always < 16*65 = 1040
    const int e1 = tid + kBLK;
    const int m0 = e0 / kH, j0 = e0 - m0 * kH;
    const int m1 = e1 / kH, j1 = e1 - m1 * kH;
    float c0 = 0.0f, c1 = 0.0f;

    __syncthreads();

    for (int t = 0; t < kT; ++t) {
        // ---- GEMM phase: acc = pre_t(tile) + h @ Whh(tile) ----
        const float* pbase = pre + ((size_t)t * kB + b0) * (size_t)kG;
        v8f acc;
#pragma unroll
        for (int r = 0; r < 8; ++r) {
            int m = r + mhi;
            acc[r] = (g < kG) ? pbase[(size_t)m * kG + g] : 0.0f;
        }
        if ((t + 1 < kT) && (g < kG)) {
            __builtin_prefetch(pre + ((size_t)(t + 1) * kB + b0 + mhi) * (size_t)kG + g, 0, 0);
        }
#pragma unroll
        for (int kc = 0; kc < kNT; ++kc) {
            int kb = (kc << 2) + khalf;
            v2f a = *(const v2f*)&shH[n * kKP + kb];
            acc = __builtin_amdgcn_wmma_f32_16x16x4_f32(
                      false, a, false, wreg[kc], (short)0, acc, false, false);
        }
#pragma unroll
        for (int r = 0; r < 8; ++r) {
            shG[(r + mhi) * kGP + g] = acc[r];
        }
        __syncthreads();

        // ---- elementwise gate phase ----
        {
            float gi = sigm(shG[m0 * kGP + j0]);
            float gf = sigm(shG[m0 * kGP + kH     + j0]);
            float gg = tanh_fast(shG[m0 * kGP + 2 * kH + j0]);
            float go = sigm(shG[m0 * kGP + 3 * kH + j0]);
            c0 = gf * c0 + gi * gg;
            float h = go * tanh_fast(c0);
            shH[m0 * kKP + j0] = h;
            if (h_all)
                h_all[(size_t)(b0 + m0) * (size_t)(kT * kH) + (size_t)t * kH + j0] = h;
            if (h_last && t == kT - 1)
                h_last[(b0 + m0) * kH + j0] = h;
        }
        if (e1 < 16 * kH) {
            float gi = sigm(shG[m1 * kGP + j1]);
            float gf = sigm(shG[m1 * kGP + kH     + j1]);
            float gg = tanh_fast(shG[m1 * kGP + 2 * kH + j1]);
            float go = sigm(shG[m1 * kGP + 3 * kH + j1]);
            c1 = gf * c1 + gi * gg;
            float h = go * tanh_fast(c1);
            shH[m1 * kKP + j1] = h;
            if (h_all)
                h_all[(size_t)(b0 + m1) * (size_t)(kT * kH) + (size_t)t * kH + j1] = h;
            if (h_last && t == kT - 1)
                h_last[(b0 + m1) * kH + j1] = h;
        }
        __syncthreads();
    }
}

// ---------------------------------------------------------------------------
// Kernel 3: out[b] = h2_last[b] . w_lin + b_lin
// ---------------------------------------------------------------------------
__global__ void final_linear(const float* __restrict__ h_last,
                             const float* __restrict__ w_lin,
                             const float* __restrict__ b_lin,
                             float* __restrict__ out)
{
    int b = blockIdx.x * blockDim.x + threadIdx.x;
    if (b < kB) {
        float s = 0.0f;
#pragma unroll
        for (int j = 0; j < kH; ++j) s += h_last[b * kH + j] * w_lin[j];
        out[b] = s + b_lin[0];
    }
}

// ---------------------------------------------------------------------------
extern "C" void kernel_launch(void* const* d_in, const int* in_sizes, int n_in,
                              void* d_out, int out_size, void* d_ws, size_t ws_size,
                              hipStream_t stream)
{
    (void)in_sizes; (void)n_in; (void)out_size; (void)ws_size;

    const float* x     = (const float*)d_in[0];
    const float* w_ih0 = (const float*)d_in[1];
    const float* w_hh0 = (const float*)d_in[2];
    const float* b_ih0 = (const float*)d_in[3];
    const float* b_hh0 = (const float*)d_in[4];
    const float* w_ih1 = (const float*)d_in[5];
    const float* w_hh1 = (const float*)d_in[6];
    const float* b_ih1 = (const float*)d_in[7];
    const float* b_hh1 = (const float*)d_in[8];
    const float* w_lin = (const float*)d_in[9];
    const float* b_lin = (const float*)d_in[10];
    float* out = (float*)d_out;

    char* ws = (char*)d_ws;
    const size_t preBytes = (size_t)kT * kB * kG * sizeof(float);   // 545 MB
    const size_t h1Bytes  = (size_t)kB * kT * kH * sizeof(float);   // 136 MB
    float* pre    = (float*)ws;
    float* h1     = (float*)(ws + preBytes);
    float* h2last = (float*)(ws + preBytes + h1Bytes);

    dim3 gemmGrid(kT * (kB / 16));   // 32768 blocks

    // Layer 0
    lstm_pre_gemm<<<gemmGrid, kBLK, 0, stream>>>(x, w_ih0, b_ih0, b_hh0, pre);
    lstm_scan<<<kB / 16, kBLK, 0, stream>>>(pre, w_hh0, h1, nullptr);
    // Layer 1 (reuses the pre buffer)
    lstm_pre_gemm<<<gemmGrid, kBLK, 0, stream>>>(h1, w_ih1, b_ih1, b_hh1, pre);
    lstm_scan<<<kB / 16, kBLK, 0, stream>>>(pre, w_hh1, nullptr, h2last);
    // Final projection
    final_linear<<<2, 256, 0, stream>>>(h2last, w_lin, b_lin, out);
}